// ROIAlign_64690797412515
// MI455X (gfx1250) — compile-verified
//
#include <hip/hip_runtime.h>
#include <stdint.h>

// ---------------- problem constants (match reference) ----------------
#define OUT_H 7
#define OUT_W 7
#define NBINS (OUT_H * OUT_W)      // 49
#define SRATIO 2                   // sampling_ratio
#define NS (OUT_H * SRATIO)        // 14 sample coords per axis
#define C_TOTAL 256
#define CT 32                      // channels per TDM tile
#define NT (C_TOTAL / CT)          // 8 tiles per ROI
#define FH 128
#define FW 128
#define FHW (FH * FW)
#define SPATIAL_SCALE 0.25f

// Max padded patch: roi spans <=32 feature px -> <=34 cells, padded to odd 35.
#define MAXD 35
#define BUF_ELEMS (CT * MAXD * MAXD)          // 39200 floats
#define BUF_BYTES (BUF_ELEMS * 4)             // 156800 B
#define DYN_SHMEM (2 * BUF_BYTES)             // 313600 B (two TDM buffers)

typedef uint32_t u32x4 __attribute__((ext_vector_type(4)));
typedef int      i32x4 __attribute__((ext_vector_type(4)));
typedef int      i32x8 __attribute__((ext_vector_type(8)));

__device__ __forceinline__ int imin(int a, int b) { return a < b ? a : b; }

extern "C" __global__ void __launch_bounds__(256, 1)
roi_align_tdm_kernel(const float* __restrict__ feat,
                     const float* __restrict__ rois,
                     float* __restrict__ out)
{
    extern __shared__ float dynbuf[];   // 2 x BUF_ELEMS, TDM destination

    // per-ROI sampling tables (static LDS, tiny)
    __shared__ int   s_yl[NS], s_yh[NS], s_xl[NS], s_xh[NS];
    __shared__ float s_wy0[NS], s_wy1[NS], s_wx0[NS], s_wx1[NS];

    const int roi = blockIdx.x;
    const int tid = threadIdx.x;

    // ---- ROI decode (uniform; every lane computes the same values) ----
    const float* r = rois + (size_t)roi * 5;
    const int   b  = (int)r[0];
    const float x1 = r[1] * SPATIAL_SCALE;
    const float y1 = r[2] * SPATIAL_SCALE;
    const float x2 = r[3] * SPATIAL_SCALE;
    const float y2 = r[4] * SPATIAL_SCALE;
    const float roi_w = fmaxf(x2 - x1, 1.0f);
    const float roi_h = fmaxf(y2 - y1, 1.0f);
    const float bin_w = roi_w * (1.0f / OUT_W);
    const float bin_h = roi_h * (1.0f / OUT_H);

    // ---- sampling grid setup: thread i < 14 handles sample i on both axes ----
    if (tid < NS) {
        const int i = tid;
        const float t = (float)(i >> 1) + ((float)(i & 1) + 0.5f) * 0.5f;
        { // y axis
            const float c  = y1 + bin_h * t;
            const bool  ok = (c >= -1.0f) && (c <= (float)FH);
            const float cc = fmaxf(c, 0.0f);
            const int   lo = imin((int)floorf(cc), FH - 1);
            const int   hi = imin(lo + 1, FH - 1);
            const float fr = cc - (float)lo;
            s_yl[i] = lo; s_yh[i] = hi;
            s_wy0[i] = ok ? (1.0f - fr) : 0.0f;
            s_wy1[i] = ok ? fr : 0.0f;
        }
        { // x axis
            const float c  = x1 + bin_w * t;
            const bool  ok = (c >= -1.0f) && (c <= (float)FW);
            const float cc = fmaxf(c, 0.0f);
            const int   lo = imin((int)floorf(cc), FW - 1);
            const int   hi = imin(lo + 1, FW - 1);
            const float fr = cc - (float)lo;
            s_xl[i] = lo; s_xh[i] = hi;
            s_wx0[i] = ok ? (1.0f - fr) : 0.0f;
            s_wx1[i] = ok ? fr : 0.0f;
        }
    }
    __syncthreads();

    // ---- uniform patch bounds (sample coords are monotonic) ----
    const int y_lo = s_yl[0], y_hi = s_yh[NS - 1];
    const int x_lo = s_xl[0], x_hi = s_xh[NS - 1];
    const int HrP = ((y_hi - y_lo + 1) | 1);   // odd-padded tile dims ->
    const int WrP = ((x_hi - x_lo + 1) | 1);   // odd channel stride in LDS
    const int chanStride = HrP * WrP;

    const uint32_t ldsBase = (uint32_t)(uintptr_t)&dynbuf[0]; // flat LDS low32 = LDS offset

    // ---- TDM issue: one 3-D tile (WrP x HrP x CT) of the NCHW tensor -> LDS ----
    auto issue_tdm = [&](int tile, int parity) {
        const int c0 = tile * CT;
        const uint64_t gaddr = (uint64_t)(uintptr_t)feat +
            (uint64_t)(((size_t)(b * C_TOTAL + c0) * FHW) + (size_t)y_lo * FW + (size_t)x_lo) * 4ull;
        const uint32_t lds_addr = ldsBase + (uint32_t)parity * (uint32_t)BUF_BYTES;
        const uint32_t td0 = (uint32_t)(FW - x_lo);       // remaining extent -> HW OOB zero-fill
        const uint32_t td1 = (uint32_t)(FH - y_lo);

        u32x4 g0; i32x8 g1; i32x4 g2; i32x4 g3; i32x8 g4;
        // group0: count=1 | lds_addr | global_addr[56:0] | type=2
        g0[0] = 1u;
        g0[1] = lds_addr;
        g0[2] = (uint32_t)gaddr;
        g0[3] = ((uint32_t)(gaddr >> 32) & 0x01FFFFFFu) | (2u << 30);
        // group1 (256-bit)
        g1[0] = (int)(2u << 16);                                 // data_size = 4B; no multicast
        g1[1] = (int)((td0 & 0xFFFFu) << 16);                    // tensor_dim0[15:0]
        g1[2] = (int)(((td0 >> 16) & 0xFFFFu) | ((td1 & 0xFFFFu) << 16));
        g1[3] = (int)(((td1 >> 16) & 0xFFFFu) | ((uint32_t)WrP << 16)); // tile_dim0
        g1[4] = (int)((uint32_t)HrP | ((uint32_t)CT << 16));     // tile_dim1 | tile_dim2
        g1[5] = (int)(uint32_t)FW;                               // tensor_dim0_stride (elements)
        g1[6] = (int)(((uint32_t)FHW & 0xFFFFu) << 16);          // tensor_dim1_stride[15:0]
        g1[7] = (int)((uint32_t)FHW >> 16);                      // tensor_dim1_stride[47:16]
        // group2: tensor_dim2 (channel extent); 3-D tile, no iteration
        g2[0] = (int)(uint32_t)(C_TOTAL - c0);
        g2[1] = 0; g2[2] = 0; g2[3] = 0;
        // group3: unused higher dims
        g3[0] = 0; g3[1] = 0; g3[2] = 0; g3[3] = 0;
        // trailing 8-dword group (gather/VADDR4 slot on clang-23): unused
        g4[0] = 0; g4[1] = 0; g4[2] = 0; g4[3] = 0;
        g4[4] = 0; g4[5] = 0; g4[6] = 0; g4[7] = 0;

        // amdgpu-toolchain (clang-23) 6-arg form; lowers to tensor_load_to_lds
        __builtin_amdgcn_tensor_load_to_lds(g0, g1, g2, g3, g4, 0);
    };

    const int wave     = tid >> 5;
    const int c_local  = tid & (CT - 1);   // lane -> channel (odd LDS stride, no bank conflicts)
    const int binStart = tid >> 5;         // wave -> bin slice

    if (wave == 0) issue_tdm(0, 0);

    float* outRoi = out + (size_t)roi * C_TOTAL * NBINS;

    for (int t = 0; t < NT; ++t) {
        if (wave == 0) {
            if (t + 1 < NT) {
                issue_tdm(t + 1, (t + 1) & 1);          // prefetch next tile (double buffer)
                __builtin_amdgcn_s_wait_tensorcnt(1);   // tile t complete
            } else {
                __builtin_amdgcn_s_wait_tensorcnt(0);
            }
        }
        __syncthreads();   // publish tile t LDS data to all waves

        const float* buf = dynbuf + (size_t)(t & 1) * BUF_ELEMS + (size_t)c_local * chanStride;

        for (int bin = binStart; bin < NBINS; bin += 8) {
            const int py = bin / OUT_W;
            const int px = bin - py * OUT_W;
            float acc = 0.0f;
#pragma unroll
            for (int sy = 0; sy < SRATIO; ++sy) {
                const int   iy  = py * SRATIO + sy;
                const int   yl  = s_yl[iy] - y_lo;
                const int   yh  = s_yh[iy] - y_lo;
                const float wy0 = s_wy0[iy];
                const float wy1 = s_wy1[iy];
                const float* row0 = buf + yl * WrP;
                const float* row1 = buf + yh * WrP;
#pragma unroll
                for (int sx = 0; sx < SRATIO; ++sx) {
                    const int   ix  = px * SRATIO + sx;
                    const int   xl  = s_xl[ix] - x_lo;
                    const int   xh  = s_xh[ix] - x_lo;
                    const float wx0 = s_wx0[ix];
                    const float wx1 = s_wx1[ix];
                    acc += wy0 * (wx0 * row0[xl] + wx1 * row0[xh])
                         + wy1 * (wx0 * row1[xl] + wx1 * row1[xh]);
                }
            }
            outRoi[(size_t)(t * CT + c_local) * NBINS + bin] = acc * 0.25f;
        }
        __syncthreads();   // all reads of buf[t&1] done before it is reused as dest
    }
}

extern "C" void kernel_launch(void* const* d_in, const int* in_sizes, int n_in,
                              void* d_out, int out_size, void* d_ws, size_t ws_size,
                              hipStream_t stream) {
    (void)n_in; (void)out_size; (void)d_ws; (void)ws_size;
    const float* feat = (const float*)d_in[0];
    const float* rois = (const float*)d_in[1];
    float* out = (float*)d_out;

    const int K = in_sizes[1] / 5;   // 1024 ROIs

    // allow 313 KB dynamic LDS (CDNA5 WGP supports up to 320 KB per workgroup)
    (void)hipFuncSetAttribute((const void*)roi_align_tdm_kernel,
                              hipFuncAttributeMaxDynamicSharedMemorySize, DYN_SHMEM);

    roi_align_tdm_kernel<<<dim3(K), dim3(256), DYN_SHMEM, stream>>>(feat, rois, out);
}